// NMSLayer_82136954568858
// MI455X (gfx1250) — compile-verified
//
#include <hip/hip_runtime.h>
#include <stdint.h>

// Greedy NMS, N=262144, max_out=400, iou_thr=0.5.
// init + 400 fused (suppress-by-previous-winner + argmax) rounds + finalize.
// L2-resident working set (5MB << 192MB). CDNA5 paths: batched async global->LDS
// (ASYNCcnt + s_wait_asynccnt), global_atomic_max_u64 tournament, LDS reduce.

typedef unsigned long long u64;
typedef uint32_t u32;

#define MAX_OUT 400
#define BLOCK 256
#define ELEMS 4          // elements per thread -> 256 blocks per round

// ---- CDNA5 async global->LDS copy of one 16B box per lane --------------------
// Verified signature: (int4 AS1* src, int4 AS3* dst, imm offset, imm cpol)
typedef int v4i __attribute__((vector_size(16)));

__device__ __forceinline__ void copy_box_to_lds(const float* g, float* l) {
#if defined(__gfx1250__) && __has_builtin(__builtin_amdgcn_global_load_async_to_lds_b128)
  typedef __attribute__((address_space(1))) v4i AS1v4i;
  typedef __attribute__((address_space(3))) v4i AS3v4i;
  __builtin_amdgcn_global_load_async_to_lds_b128(
      (AS1v4i*)(uintptr_t)g,
      (AS3v4i*)(u32)(uintptr_t)l,
      /*offset=*/0, /*cpol=*/0);
#else
  *(float4*)l = *(const float4*)g;
#endif
}

__device__ __forceinline__ void wait_async_zero() {
#if defined(__gfx1250__) && __has_builtin(__builtin_amdgcn_s_wait_asynccnt)
  __builtin_amdgcn_s_wait_asynccnt(0);
#elif defined(__gfx1250__)
  asm volatile("s_wait_asynccnt 0x0" ::: "memory");
#endif
}

// Pack (score, idx) so that u64-max picks max score, ties -> smallest idx.
// Scores are in [0,1) (non-negative) so raw float bits are order-preserving.
__device__ __forceinline__ u64 pack_key(float s, u32 idx) {
  return (((u64)__float_as_uint(s)) << 32) | (u64)(~idx);
}

// ---- round kernel (defined first so disasm snippet shows the async path) -----
__global__ __launch_bounds__(BLOCK) void nms_round(const float* __restrict__ boxes,
                                                   float* __restrict__ wscore,
                                                   u64* __restrict__ slots,
                                                   int r, int n) {
  __shared__ __align__(16) float lds_box[BLOCK * ELEMS * 4];
  __shared__ u64 red[BLOCK];

  const int tid  = threadIdx.x;
  const int base = blockIdx.x * (BLOCK * ELEMS) + tid;   // + e*BLOCK, coalesced

  // Load my ELEMS scores.
  float s[ELEMS];
  bool act[ELEMS];
#pragma unroll
  for (int e = 0; e < ELEMS; ++e) {
    int gid = base + e * BLOCK;
    s[e] = (gid < n) ? wscore[gid] : -INFINITY;
    act[e] = (gid < n) && (s[e] > -INFINITY);
  }

  // Fetch previous round's winner (wave-uniform).
  bool have_w = false;
  u32 widx = 0u;
  float wy1 = 0.f, wx1 = 0.f, wy2 = 0.f, wx2 = 0.f, warea = 0.f;
  if (r > 0) {
    u64 w = slots[r - 1];
    if (w != 0ull) {
      widx = ~(u32)w;
      float4 wb = *(const float4*)(boxes + 4ull * widx);
      wy1 = fminf(wb.x, wb.z); wy2 = fmaxf(wb.x, wb.z);
      wx1 = fminf(wb.y, wb.w); wx2 = fmaxf(wb.y, wb.w);
      warea = (wy2 - wy1) * (wx2 - wx1);
      have_w = true;
    }
  }

  if (have_w) {
    // Issue all async box fetches (EXEC-predicated on still-active), then one wait.
#pragma unroll
    for (int e = 0; e < ELEMS; ++e) {
      if (act[e]) {
        int gid = base + e * BLOCK;
        copy_box_to_lds(boxes + 4ull * gid, &lds_box[4 * (e * BLOCK + tid)]);
      }
    }
    wait_async_zero();
    // Consume: IoU vs winner, suppress.
#pragma unroll
    for (int e = 0; e < ELEMS; ++e) {
      if (act[e]) {
        int gid = base + e * BLOCK;
        float4 b = *(float4*)&lds_box[4 * (e * BLOCK + tid)];
        float y1 = fminf(b.x, b.z), y2 = fmaxf(b.x, b.z);
        float x1 = fminf(b.y, b.w), x2 = fmaxf(b.y, b.w);
        float area = (y2 - y1) * (x2 - x1);
        float yy1 = fmaxf(y1, wy1), xx1 = fmaxf(x1, wx1);
        float yy2 = fminf(y2, wy2), xx2 = fminf(x2, wx2);
        float inter = fmaxf(yy2 - yy1, 0.f) * fmaxf(xx2 - xx1, 0.f);
        float uni = area + warea - inter;
        float iou = (inter > 0.f) ? (inter / uni) : 0.f;
        if (iou > 0.5f || gid == (int)widx) {
          s[e] = -INFINITY;
          act[e] = false;
          wscore[gid] = -INFINITY;
        }
      }
    }
  }

  // Thread-local argmax over my elements, then block reduce, then 1 atomic.
  u64 key = 0ull;
#pragma unroll
  for (int e = 0; e < ELEMS; ++e) {
    if (act[e]) {
      u64 k = pack_key(s[e], (u32)(base + e * BLOCK));
      if (k > key) key = k;
    }
  }
  red[tid] = key;
  __syncthreads();
#pragma unroll
  for (int off = BLOCK / 2; off > 0; off >>= 1) {
    if (tid < off) {
      u64 o = red[tid + off];
      if (o > red[tid]) red[tid] = o;
    }
    __syncthreads();
  }
  if (tid == 0 && red[0] != 0ull) {
    atomicMax(&slots[r], red[0]);
  }
}

// ---- other kernels -----------------------------------------------------------
__global__ __launch_bounds__(BLOCK) void nms_init(const float* __restrict__ scores,
                                                  float* __restrict__ wscore,
                                                  u64* __restrict__ slots, int n) {
  int gid = blockIdx.x * BLOCK + threadIdx.x;
  if (gid < n) wscore[gid] = scores[gid];
  if (gid < MAX_OUT) slots[gid] = 0ull;
}

__global__ void nms_finalize(const u64* __restrict__ slots, int* __restrict__ out) {
  __shared__ int cnt;
  int t = threadIdx.x;
  if (t == 0) cnt = 0;
  __syncthreads();
  if (t < MAX_OUT) {
    u64 s = slots[t];
    bool ok = (s != 0ull);
    out[t] = ok ? (int)(~(u32)s) : 0;
    if (ok) atomicAdd(&cnt, 1);
  }
  __syncthreads();
  if (t == 0) out[MAX_OUT] = cnt;
}

// ---- entry -------------------------------------------------------------------
extern "C" void kernel_launch(void* const* d_in, const int* in_sizes, int n_in,
                              void* d_out, int out_size, void* d_ws, size_t ws_size,
                              hipStream_t stream) {
  const float* boxes  = (const float*)d_in[0];   // [N,4] fp32
  const float* scores = (const float*)d_in[1];   // [N]   fp32
  const int n = in_sizes[1];

  float* wscore = (float*)d_ws;
  size_t slot_off = ((size_t)n * sizeof(float) + 255) & ~(size_t)255;
  u64* slots = (u64*)((char*)d_ws + slot_off);

  const int nblk_init  = (n + BLOCK - 1) / BLOCK;
  const int nblk_round = (n + BLOCK * ELEMS - 1) / (BLOCK * ELEMS);

  nms_init<<<nblk_init, BLOCK, 0, stream>>>(scores, wscore, slots, n);
  for (int r = 0; r < MAX_OUT; ++r) {
    nms_round<<<nblk_round, BLOCK, 0, stream>>>(boxes, wscore, slots, r, n);
  }
  nms_finalize<<<1, 512, 0, stream>>>(slots, (int*)d_out);
}